// KalmanFilter_592705487335
// MI455X (gfx1250) — compile-verified
//
#include <hip/hip_runtime.h>
#include <hip/hip_bf16.h>

typedef float v2f __attribute__((ext_vector_type(2)));
typedef float v8f __attribute__((ext_vector_type(8)));

#define T_STEPS 128
#define BATCH   4096
#define SD      16
#define OD      8
#define WPB     4          // waves per block
#define TILE_F  256        // floats per 16x16 tile
#define NTILES  6

// D = A(16x4) * B(4x16) + C   (fp32 WMMA, wave32)
__device__ __forceinline__ v8f wmma4(v2f a, v2f b, v8f c) {
  return __builtin_amdgcn_wmma_f32_16x16x4_f32(false, a, false, b, (short)0, c,
                                               false, false);
}

// ---- layout helpers -------------------------------------------------------
// C/D layout: reg r, lane l -> element (M = r + 8*(l>=16), N = l&15)
// A layout  : reg v, lane l -> element (M = l&15, K = v + 2*(l>=16))
// B layout  : reg v, lane l -> element (K = v + 2*(l>=16), N = l&15)
__device__ __forceinline__ void store_c(float* t, v8f c, int n, int h) {
#pragma unroll
  for (int r = 0; r < 8; ++r) t[(r + 8 * h) * 16 + n] = c[r];
}
__device__ __forceinline__ v2f load_a(const float* t, int kc, int m, int h) {
  v2f a;
  a.x = t[m * 16 + kc + 2 * h];
  a.y = t[m * 16 + kc + 2 * h + 1];
  return a;
}
__device__ __forceinline__ v2f load_b(const float* t, int kc, int n, int h) {
  v2f b;
  b.x = t[(kc + 2 * h) * 16 + n];
  b.y = t[(kc + 2 * h + 1) * 16 + n];
  return b;
}
// D = Aconst(16x16) * Btile + C  (chain of 4 K=4 WMMAs)
__device__ __forceinline__ v8f chain_constA(const v2f A[4], const float* tB,
                                            v8f c, int n, int h) {
#pragma unroll
  for (int kc = 0; kc < 16; kc += 4)
    c = wmma4(A[kc >> 2], load_b(tB, kc, n, h), c);
  return c;
}
// D = Atile * Bconst(16x16) + C
__device__ __forceinline__ v8f chain_constB(const float* tA, const v2f Bv[4],
                                            v8f c, int m, int h) {
#pragma unroll
  for (int kc = 0; kc < 16; kc += 4)
    c = wmma4(load_a(tA, kc, m, h), Bv[kc >> 2], c);
  return c;
}

__global__ __launch_bounds__(WPB * 32) void kalman_wmma_kernel(
    const float* __restrict__ obs,  // (T,B,8)
    const float* __restrict__ Fm,   // (16,16)
    const float* __restrict__ Hm,   // (8,16)
    const float* __restrict__ Qm,   // (16,16)
    const float* __restrict__ Rm,   // (8,8)
    const float* __restrict__ m0g,  // (B,16)
    const float* __restrict__ P0g,  // (B,16,16)
    float* __restrict__ outM,       // (T,B,16)
    float* __restrict__ outP)       // (T,B,16,16)
{
  __shared__ float lds[WPB * NTILES * TILE_F];

  const int lane = threadIdx.x & 31;
  const int wid  = threadIdx.x >> 5;
  const int b    = blockIdx.x * WPB + wid;   // one wave per batch element
  const int n    = lane & 15;                // C/B-layout column
  const int h    = lane >> 4;                // half (0: lanes 0-15, 1: 16-31)
  const int m    = n;                        // A-layout row

  float* tP   = &lds[(wid * NTILES + 0) * TILE_F];
  float* tHP  = &lds[(wid * NTILES + 1) * TILE_F];
  float* tMS  = &lds[(wid * NTILES + 2) * TILE_F];  // Mmat, later S
  float* tPHT = &lds[(wid * NTILES + 3) * TILE_F];
  float* tK   = &lds[(wid * NTILES + 4) * TILE_F];
  float* tSI  = &lds[(wid * NTILES + 5) * TILE_F];

  // ---- wave-resident constants --------------------------------------------
  // hAB: padded H in A layout == padded H^T in B layout (same per-lane values)
  // fa : F in A layout == F^T in B layout
  v2f hAB[4], fa[4];
#pragma unroll
  for (int kc = 0; kc < 16; kc += 4) {
    int k0 = kc + 2 * h;
    v2f hv;
    hv.x = (m < OD) ? Hm[m * SD + k0] : 0.f;
    hv.y = (m < OD) ? Hm[m * SD + k0 + 1] : 0.f;
    hAB[kc >> 2] = hv;
    v2f fv;
    fv.x = Fm[m * SD + k0];
    fv.y = Fm[m * SD + k0 + 1];
    fa[kc >> 2] = fv;
  }
  v8f qc;
#pragma unroll
  for (int r = 0; r < 8; ++r) qc[r] = Qm[(r + 8 * h) * SD + n];

  // Gauss-Jordan lane mapping: row gi = lane&7, col-group gj = lane>>3
  const int gi = lane & 7, gj = lane >> 3;
  float rreg[4];
#pragma unroll
  for (int c = 0; c < 4; ++c)
    rreg[c] = (gj < 2) ? Rm[gi * OD + gj * 4 + c] : 0.f;

  // tSI padding (rows>=8 / cols>=8) stays zero for the whole kernel
  for (int idx = lane; idx < TILE_F; idx += 32) tSI[idx] = 0.f;

  // ---- initial state ------------------------------------------------------
  for (int idx = lane; idx < TILE_F; idx += 32)
    tP[idx] = P0g[(size_t)b * TILE_F + idx];
  for (int idx = lane; idx < TILE_F; idx += 32)
    tMS[idx] = m0g[b * SD + (idx >> 4)];   // column-replicated mean matrix
  __builtin_amdgcn_wave_barrier();

  // initial predict: P = F P0 F^T + Q ; Mmat = F Mmat0
  v8f z = {};
  v8f fp = chain_constA(fa, tP, z, n, h);
  v8f mm = chain_constA(fa, tMS, z, n, h);
  store_c(tHP, fp, n, h);
  __builtin_amdgcn_wave_barrier();
  v8f p = chain_constB(tHP, fa, qc, m, h);  // fa doubles as F^T B-operand

  // ---- sequential scan over T --------------------------------------------
  for (int t = 0; t < T_STEPS; ++t) {
    // prefetch next step's observations (one line per wave)
    if (t + 1 < T_STEPS && lane == 0)
      __builtin_prefetch(&obs[((size_t)(t + 1) * BATCH + b) * OD], 0, 1);

    store_c(tP, p, n, h);
    store_c(tMS, mm, n, h);
    __builtin_amdgcn_wave_barrier();

    v8f hp = chain_constA(hAB, tP, z, n, h);   // H P   (rows 8..15 == 0)
    v8f hm = chain_constA(hAB, tMS, z, n, h);  // H m, column-replicated
    store_c(tHP, hp, n, h);
    __builtin_amdgcn_wave_barrier();

    v8f sC = chain_constB(tHP, hAB, z, m, h);  // (H P) H^T
    store_c(tMS, sC, n, h);
    __builtin_amdgcn_wave_barrier();

    // ---- 8x8 Gauss-Jordan inverse of S = HPH^T + R (SPD, no pivoting) ----
    float g[4];
#pragma unroll
    for (int c = 0; c < 4; ++c) {
      int j = gj * 4 + c;                      // augmented column 0..15
      g[c] = (gj < 2) ? (tMS[gi * 16 + j] + rreg[c])
                      : ((gi == j - 8) ? 1.f : 0.f);
    }
#pragma unroll
    for (int k = 0; k < 8; ++k) {
      float piv = __shfl(g[k & 3], (k >> 2) * 8 + k, 32);
      float rp  = 1.f / piv;
#pragma unroll
      for (int c = 0; c < 4; ++c) {
        float sc = g[c] * rp;
        g[c] = (gi == k) ? sc : g[c];          // scale pivot row
      }
      float f = __shfl(g[k & 3], (k >> 2) * 8 + gi, 32);  // A[i][k]
      float akj[4];
#pragma unroll
      for (int c = 0; c < 4; ++c) akj[c] = __shfl(g[c], gj * 8 + k, 32);
#pragma unroll
      for (int c = 0; c < 4; ++c) {
        float e = g[c] - f * akj[c];
        g[c] = (gi == k) ? g[c] : e;           // eliminate other rows
      }
    }
#pragma unroll
    for (int c = 0; c < 4; ++c)
      if (gj >= 2) tSI[gi * 16 + (gj - 2) * 4 + c] = g[c];  // top-left 8x8
    __builtin_amdgcn_wave_barrier();

    // PHt = P H^T  (cols 8..15 == 0)
    v8f pht = chain_constB(tP, hAB, z, m, h);
    store_c(tPHT, pht, n, h);
    __builtin_amdgcn_wave_barrier();

    // K = PHt * Sinv : only K-dim chunks 0,4 are nonzero
    v8f kk = z;
    kk = wmma4(load_a(tPHT, 0, m, h), load_b(tSI, 0, n, h), kk);
    kk = wmma4(load_a(tPHT, 4, m, h), load_b(tSI, 4, n, h), kk);
    v8f kneg = -kk;                            // store -K (f32 WMMA has no A-neg)
    store_c(tK, kneg, n, h);
    __builtin_amdgcn_wave_barrier();

    // negated residual (Hm - y), column-replicated, rows 8..15 == 0
    float yv = (lane < OD) ? obs[((size_t)t * BATCH + b) * OD + lane] : 0.f;
    float rm[8];
#pragma unroll
    for (int r = 0; r < 8; ++r) {
      float yb = __shfl(yv, r, 32);
      rm[r] = hm[r] - ((h == 0) ? yb : 0.f);
    }
    float rr[8];
#pragma unroll
    for (int r = 0; r < 8; ++r) rr[r] = __shfl(rm[r], 0, 32);
    v2f rb0, rb4;                              // resid * 1^T in B layout
    rb0.x = h ? rr[2] : rr[0];  rb0.y = h ? rr[3] : rr[1];
    rb4.x = h ? rr[6] : rr[4];  rb4.y = h ? rr[7] : rr[5];

    v2f ak0 = load_a(tK, 0, m, h),  ak4 = load_a(tK, 4, m, h);
    v2f hb0 = load_b(tHP, 0, n, h), hb4 = load_b(tHP, 4, n, h);

    mm = wmma4(ak0, rb0, mm);   // M += (-K)(Hm - y) == M + K(y - Hm)
    mm = wmma4(ak4, rb4, mm);
    p  = wmma4(ak0, hb0, p);    // P -= K (H P)
    p  = wmma4(ak4, hb4, p);

    // ---- outputs: streamed once, bypass temporal caching (TH=NT) ---------
    float mv = 0.f;
#pragma unroll
    for (int r = 0; r < 8; ++r) {
      float hiv  = __shfl(mm[r], 16, 32);      // m[r+8] (column-replicated)
      float cand = (lane & 8) ? hiv : mm[r];
      mv = ((lane & 7) == r) ? cand : mv;
    }
    if (lane < SD)
      __builtin_nontemporal_store(mv, &outM[((size_t)t * BATCH + b) * SD + lane]);

    size_t cb = ((size_t)t * BATCH + b) * (size_t)TILE_F;
#pragma unroll
    for (int r = 0; r < 8; ++r)
      __builtin_nontemporal_store(p[r], &outP[cb + (r + 8 * h) * 16 + n]);
  }
}

extern "C" void kernel_launch(void* const* d_in, const int* in_sizes, int n_in,
                              void* d_out, int out_size, void* d_ws,
                              size_t ws_size, hipStream_t stream) {
  const float* obs = (const float*)d_in[0];
  const float* F   = (const float*)d_in[1];
  const float* H   = (const float*)d_in[2];
  const float* Q   = (const float*)d_in[3];
  const float* R   = (const float*)d_in[4];
  const float* m0  = (const float*)d_in[5];
  const float* P0  = (const float*)d_in[6];
  float* outM = (float*)d_out;
  float* outP = outM + (size_t)T_STEPS * BATCH * SD;

  dim3 block(WPB * 32);
  dim3 grid(BATCH / WPB);
  hipLaunchKernelGGL(kalman_wmma_kernel, grid, block, 0, stream,
                     obs, F, H, Q, R, m0, P0, outM, outP);
}